// RelationalAttention_38646115729913
// MI455X (gfx1250) — compile-verified
//
#include <hip/hip_runtime.h>
#include <hip/hip_bf16.h>

typedef float v2f __attribute__((ext_vector_type(2)));
typedef float v8f __attribute__((ext_vector_type(8)));

#define BT      128
#define NTOK    32
#define HEADS   8
#define DH      32
#define DMODEL  256
#define QKVLD   768
#define LDSS    260   // 256 + 4 pad floats: rows stay 16B aligned for async b128;
                      // 260 mod 64 = 4 -> worst case 2-way bank conflict (score phase),
                      // output phase (consecutive d) conflict-free.

// ---------------------------------------------------------------------------
// CDNA5 async global->LDS copy (bypasses VGPRs, tracked by ASYNCcnt).
// VDST operand carries the workgroup-relative LDS byte offset, which is the
// low 32 bits of a generic pointer into shared memory (shared-aperture flat
// address = {aperture_hi32, lds_offset_lo32}).
// ---------------------------------------------------------------------------
__device__ __forceinline__ void async_load_b128_to_lds(void* lds_dst, const void* gsrc) {
  const unsigned lds_off = (unsigned)(unsigned long long)lds_dst;
  asm volatile("global_load_async_to_lds_b128 %0, %1, off"
               :: "v"(lds_off), "v"(gsrc) : "memory");
}
__device__ __forceinline__ void wait_async0() {
  asm volatile("s_wait_asynccnt 0x0" ::: "memory");
}

// ---------------------------------------------------------------------------
// Generic fp32 GEMM: C[M,N] = X[M,K] @ W[K,N] + bias[N]
// One 16x16 output tile per wave, K-loop in steps of 4 via V_WMMA_F32_16X16X4_F32.
// A frag (16x4): lane m = lane&15, ks = (lane>>4)*2 -> {A[m][ks], A[m][ks+1]}
// B frag (4x16): lane n = lane&15, ks = (lane>>4)*2 -> {B[ks][n], B[ks+1][n]}
// D (16x16 f32): VGPR r -> row (r + 8*(lane>>4)), col = lane&15
// ---------------------------------------------------------------------------
__global__ __launch_bounds__(256) void gemm_wmma_f32(
    const float* __restrict__ X, const float* __restrict__ W,
    const float* __restrict__ bias, float* __restrict__ C,
    int M, int K, int N) {
  const int lane = threadIdx.x & 31;
  const int wave = threadIdx.x >> 5;
  const int ntn  = N >> 4;
  const int tile = blockIdx.x * (blockDim.x >> 5) + wave;
  if (tile >= (M >> 4) * ntn) return;          // wave-uniform: EXEC stays all-ones
  const int mt = tile / ntn;
  const int nt = tile % ntn;

  const int mn   = lane & 15;                  // row for A frag / col for B frag
  const int half = lane >> 4;                  // 0 -> K 0,1 ; 1 -> K 2,3
  const long arow = (long)(mt * 16 + mn) * K;
  const int  bcol = nt * 16 + mn;

  v8f acc = {};
  #pragma unroll 4
  for (int k = 0; k < K; k += 4) {
    const int ks = k + half * 2;
    v2f a, b;
    a.x = X[arow + ks];
    a.y = X[arow + ks + 1];
    b.x = W[(long)ks * N + bcol];
    b.y = W[(long)(ks + 1) * N + bcol];
    acc = __builtin_amdgcn_wmma_f32_16x16x4_f32(
        /*neg_a=*/false, a, /*neg_b=*/false, b,
        /*c_mod=*/(short)0, acc, /*reuse_a=*/false, /*reuse_b=*/false);
  }

  const float bn = bias ? bias[bcol] : 0.f;
  #pragma unroll
  for (int r = 0; r < 8; ++r) {
    const int orow = mt * 16 + r + half * 8;
    C[(long)orow * N + bcol] = acc[r] + bn;
  }
}

// ---------------------------------------------------------------------------
// Fused relational attention core. One block per (bt, i); 8 waves = 8 heads.
// bias tile (32 tokens x 256 (h,d)) moved to LDS via async global->LDS b128,
// consumed twice (k_pair score phase lane=j, v_pair output phase lane=d).
// k/v rows read straight from global (8 MB total -> L2 resident).
// ---------------------------------------------------------------------------
__global__ __launch_bounds__(256) void attn_kernel(
    const float* __restrict__ qkv,          // (4096, 768) : q|k|v
    const float* __restrict__ bias,         // (4096, 32, 256)
    const unsigned char* __restrict__ mask, // (4096) bool8
    float* __restrict__ out) {              // (4096, 256)
  __shared__ float lds_bias[NTOK * LDSS];   // 33.3 KB
  __shared__ float lds_q[DMODEL];
  __shared__ float lds_attn[HEADS * NTOK];

  const int blk  = blockIdx.x;              // bt*32 + i
  const int bt   = blk >> 5;
  const int t    = threadIdx.x;
  const int lane = t & 31;
  const int h    = t >> 5;                  // wave id == head

  // stage bias tile (32 x 256 floats): 256 threads x 8 async b128 copies,
  // straight into padded LDS rows without touching VGPRs.
  const float* btile = bias + (size_t)blk * (NTOK * DMODEL);
  #pragma unroll
  for (int it = 0; it < 8; ++it) {
    const int idx = (it * 256 + t) * 4;     // element index in tile (16B granules)
    const int j = idx >> 8;
    const int c = idx & 255;
    async_load_b128_to_lds(&lds_bias[j * LDSS + c], btile + idx);
  }

  // stage q row (256 floats) while the async copies are in flight
  lds_q[t] = qkv[(size_t)blk * QKVLD + t];

  wait_async0();                            // s_wait_asynccnt 0
  __syncthreads();

  // ---- score phase: lane = token j ----
  const int j = lane;
  const float* krow = qkv + (size_t)(bt * NTOK + j) * QKVLD + DMODEL + h * DH;
  const float* bj   = &lds_bias[j * LDSS + h * DH];
  const float* qh   = &lds_q[h * DH];
  float s = 0.f;
  #pragma unroll
  for (int d = 0; d < DH; ++d)
    s += qh[d] * (krow[d] + bj[d]);
  s *= 0.17677669529663687f;                // Dh^-0.5 = 1/sqrt(32)

  const bool alive = mask[bt * NTOK + j] != 0;
  const float NEG = -3.0e38f;
  s = alive ? s : NEG;
  float mx = s;
  #pragma unroll
  for (int off = 16; off; off >>= 1)
    mx = fmaxf(mx, __shfl_xor(mx, off, 32));
  const bool row_ok = mx > 0.5f * NEG;      // false -> every token masked
  float e = (alive && row_ok) ? __expf(s - mx) : 0.f;
  float sum = e;
  #pragma unroll
  for (int off = 16; off; off >>= 1)
    sum += __shfl_xor(sum, off, 32);
  lds_attn[h * NTOK + j] = (sum > 0.f) ? (e / sum) : 0.f;
  __syncthreads();

  // ---- output phase: lane = dim d ----
  const int d = lane;
  const float* vbase = qkv + (size_t)(bt * NTOK) * QKVLD + 2 * DMODEL + h * DH + d;
  float acc = 0.f;
  #pragma unroll
  for (int jj = 0; jj < NTOK; ++jj) {
    const float w  = lds_attn[h * NTOK + jj];
    const float vp = vbase[(size_t)jj * QKVLD] + lds_bias[jj * LDSS + h * DH + d];
    acc += w * vp;
  }
  out[(size_t)blk * DMODEL + h * DH + d] = acc;
}

// ---------------------------------------------------------------------------
extern "C" void kernel_launch(void* const* d_in, const int* in_sizes, int n_in,
                              void* d_out, int out_size, void* d_ws, size_t ws_size,
                              hipStream_t stream) {
  (void)in_sizes; (void)n_in; (void)out_size; (void)ws_size;
  const float*         x      = (const float*)d_in[0];
  const float*         bias_f = (const float*)d_in[1];
  const unsigned char* mask   = (const unsigned char*)d_in[2];
  const float*         w_qkv  = (const float*)d_in[3];
  const float*         b_qkv  = (const float*)d_in[4];
  const float*         w_proj = (const float*)d_in[5];
  const float*         b_proj = (const float*)d_in[6];
  float*               out    = (float*)d_out;

  const int ROWS = BT * NTOK;               // 4096
  float* qkv_ws  = (float*)d_ws;            // 4096 * 768
  float* attn_ws = qkv_ws + (size_t)ROWS * QKVLD;   // 4096 * 256

  // 1) QKV projection: 4096x256 @ 256x768  -> 12288 tiles / 8 waves = 1536 blocks
  gemm_wmma_f32<<<1536, 256, 0, stream>>>(x, w_qkv, b_qkv, qkv_ws,
                                          ROWS, DMODEL, 3 * DMODEL);
  // 2) fused relational attention: one block per (bt, i)
  attn_kernel<<<ROWS, 256, 0, stream>>>(qkv_ws, bias_f, mask, attn_ws);
  // 3) output projection: 4096x256 @ 256x256 -> 4096 tiles / 8 = 512 blocks
  gemm_wmma_f32<<<512, 256, 0, stream>>>(attn_ws, w_proj, b_proj, out,
                                         ROWS, DMODEL, DMODEL);
}